// GCN_75677323755551
// MI455X (gfx1250) — compile-verified
//
#include <hip/hip_runtime.h>
#include <hip/hip_bf16.h>
#include <math.h>

typedef __attribute__((ext_vector_type(2))) float v2f;
typedef __attribute__((ext_vector_type(8))) float v8f;

#define NGRAPHS 512
#define DH 16

// ---------------------------------------------------------------------------
// init: deg <- 1.0 (self loop), accum <- 0, pool sums/cnts <- 0
// ---------------------------------------------------------------------------
__global__ void gcn_init(float* __restrict__ deg, float* __restrict__ accum,
                         float* __restrict__ sums, float* __restrict__ cnts,
                         int n_nodes) {
    int i = blockIdx.x * blockDim.x + threadIdx.x;
    int total = n_nodes * DH;
    if (i < total) accum[i] = 0.0f;
    if (i < n_nodes) deg[i] = 1.0f;                 // self loop contributes 1
    if (i < NGRAPHS * DH) sums[i] = 0.0f;
    if (i < NGRAPHS) cnts[i] = 0.0f;
}

// ---------------------------------------------------------------------------
// degree scatter: deg[col[e]] += 1
// ---------------------------------------------------------------------------
__global__ void gcn_degree(const int* __restrict__ ei, float* __restrict__ deg,
                           int n_edges) {
    int e = blockIdx.x * blockDim.x + threadIdx.x;
    if (e >= n_edges) return;
    int c = ei[n_edges + e];                        // edge_index[1][e]
    unsafeAtomicAdd(&deg[c], 1.0f);
}

// dinv = rsqrt(deg)  (deg >= 1 always, so no zero guard needed)
__global__ void gcn_dinv(float* __restrict__ deg, int n_nodes) {
    int v = blockIdx.x * blockDim.x + threadIdx.x;
    if (v >= n_nodes) return;
    deg[v] = rsqrtf(deg[v]);
}

// ---------------------------------------------------------------------------
// WMMA fp32 GEMM + row scale: Hs[N,16] = (X[N,KDIM] @ W[KDIM,16]) * dinv[row]
// One wave32 computes a 16-node x 16-feature tile with V_WMMA_F32_16X16X4_F32.
// A (16x4 f32): lanes 0-15 hold K=kc+0/kc+1, lanes 16-31 hold K=kc+2/kc+3.
// B (4x16 f32): same K split across lane halves, N = lane&15.
// D (16x16 f32): v8f, VGPR j -> row j (lanes 0-15) / row j+8 (lanes 16-31).
// Bounds handling is branch-free (clamped addresses + selects) so the load
// stream stays unconditional; the store path has a per-lane-half (8-row block)
// fast path (the only path taken when 16 | N).
// ---------------------------------------------------------------------------
template <int KDIM>
__global__ void gcn_gemm_wmma(const float* __restrict__ X,
                              const float* __restrict__ W,
                              const float* __restrict__ dinv,
                              float* __restrict__ Hs, int n_nodes) {
    int wave = (blockIdx.x * blockDim.x + threadIdx.x) >> 5;
    int lane = threadIdx.x & 31;
    int ntiles = (n_nodes + 15) >> 4;
    if (wave >= ntiles) return;                     // wave-uniform: EXEC all-1s inside

    const int half = lane >> 4;                     // 0: K lo pair, 1: K hi pair
    const int l15  = lane & 15;
    const int arow = wave * 16 + l15;               // A-matrix row for this lane
    const int rowc = arow < n_nodes ? arow : (n_nodes - 1);
    const float rmask = arow < n_nodes ? 1.0f : 0.0f;
    const float* __restrict__ xrow = X + (size_t)rowc * KDIM;

    v8f acc = {};
#pragma unroll
    for (int kc = 0; kc < KDIM; kc += 4) {
        const int k0 = kc + half * 2;
        const int k1 = k0 + 1;
        const int k0c = k0 < KDIM ? k0 : (KDIM - 1);
        const int k1c = k1 < KDIM ? k1 : (KDIM - 1);
        v2f a, b;
        // unconditional clamped loads; zeroing via select (v_cndmask), no branches
        float ax = xrow[k0c], ay = xrow[k1c];
        float bx = W[k0c * DH + l15], by = W[k1c * DH + l15];
        a.x = (k0 < KDIM) ? ax * rmask : 0.0f;
        a.y = (k1 < KDIM) ? ay * rmask : 0.0f;
        b.x = (k0 < KDIM) ? bx : 0.0f;
        b.y = (k1 < KDIM) ? by : 0.0f;
        acc = __builtin_amdgcn_wmma_f32_16x16x4_f32(
            /*neg_a=*/false, a, /*neg_b=*/false, b,
            /*c_mod=*/(short)0, acc, /*reuse_a=*/false, /*reuse_b=*/false);
    }

    const int rbase = wave * 16 + half * 8;         // D rows held by this lane half
    if (rbase + 8 <= n_nodes) {
        // full-tile fast path: vectorized dinv loads + unguarded store clause
        float4 d0 = *reinterpret_cast<const float4*>(dinv + rbase);
        float4 d1 = *reinterpret_cast<const float4*>(dinv + rbase + 4);
        float ds[8] = {d0.x, d0.y, d0.z, d0.w, d1.x, d1.y, d1.z, d1.w};
#pragma unroll
        for (int j = 0; j < 8; ++j)
            Hs[(size_t)(rbase + j) * DH + l15] = acc[j] * ds[j];
    } else {
#pragma unroll
        for (int j = 0; j < 8; ++j) {
            int r = rbase + j;
            if (r < n_nodes) Hs[(size_t)r * DH + l15] = acc[j] * dinv[r];
        }
    }
}

// ---------------------------------------------------------------------------
// edge scatter (pure gather + scatter-add, dinv pre-folded into Hs):
//   accum[col] += Hs[row]
// ---------------------------------------------------------------------------
__global__ void gcn_edge_agg(const int* __restrict__ ei,
                             const float* __restrict__ Hs,
                             float* __restrict__ accum, int n_edges) {
    int e = blockIdx.x * blockDim.x + threadIdx.x;
    if (e >= n_edges) return;
    int r = ei[e];
    int c = ei[n_edges + e];
    const float4* hv = reinterpret_cast<const float4*>(Hs + (size_t)r * DH);
    float* ap = accum + (size_t)c * DH;
#pragma unroll
    for (int q = 0; q < 4; ++q) {
        float4 h4 = hv[q];                           // global_load_b128 gather
        unsafeAtomicAdd(ap + q * 4 + 0, h4.x);
        unsafeAtomicAdd(ap + q * 4 + 1, h4.y);
        unsafeAtomicAdd(ap + q * 4 + 2, h4.z);
        unsafeAtomicAdd(ap + q * 4 + 3, h4.w);
    }
}

// ---------------------------------------------------------------------------
// epilogue: act = relu(dinv[v] * (accum + Hs) + b)
//   (accum holds sum of dinv[row]*h[row] over in-edges; Hs[v]=h[v]*dinv[v]
//    supplies the self loop; the outer dinv[v] completes the normalization)
// optionally re-zeroes accum for the next layer
// ---------------------------------------------------------------------------
__global__ void gcn_epilogue(const float* __restrict__ Hs,
                             float* __restrict__ accum,
                             const float* __restrict__ dinv,
                             const float* __restrict__ b,
                             float* __restrict__ act,
                             int n_nodes, int reset_accum) {
    int i = blockIdx.x * blockDim.x + threadIdx.x;
    if (i >= n_nodes * DH) return;
    int v = i >> 4;
    int f = i & 15;
    float z = dinv[v] * (accum[i] + Hs[i]) + b[f];
    act[i] = z > 0.0f ? z : 0.0f;
    if (reset_accum) accum[i] = 0.0f;
}

// ---------------------------------------------------------------------------
// mean-pool atomics per node
// ---------------------------------------------------------------------------
__global__ void gcn_pool(const float* __restrict__ act,
                         const int* __restrict__ batch,
                         float* __restrict__ sums, float* __restrict__ cnts,
                         int n_nodes) {
    int v = blockIdx.x * blockDim.x + threadIdx.x;
    if (v >= n_nodes) return;
    int g = batch[v];
    const float* a = act + (size_t)v * DH;
    float* s = sums + (size_t)g * DH;
#pragma unroll
    for (int f = 0; f < DH; ++f) unsafeAtomicAdd(s + f, a[f]);
    unsafeAtomicAdd(&cnts[g], 1.0f);
}

// head: out[g] = sigmoid( (sums[g]/max(cnt,1)) . Wl + bl )
__global__ void gcn_head(const float* __restrict__ sums,
                         const float* __restrict__ cnts,
                         const float* __restrict__ Wl,
                         const float* __restrict__ bl,
                         float* __restrict__ out, int n_graphs) {
    int g = blockIdx.x * blockDim.x + threadIdx.x;
    if (g >= n_graphs) return;
    float inv = 1.0f / fmaxf(cnts[g], 1.0f);
    float z = bl[0];
#pragma unroll
    for (int f = 0; f < DH; ++f) z += sums[(size_t)g * DH + f] * inv * Wl[f];
    out[g] = 1.0f / (1.0f + expf(-z));
}

// ---------------------------------------------------------------------------
extern "C" void kernel_launch(void* const* d_in, const int* in_sizes, int n_in,
                              void* d_out, int out_size, void* d_ws, size_t ws_size,
                              hipStream_t stream) {
    const float* x   = (const float*)d_in[0];        // [N,7]
    const int*   ei  = (const int*)d_in[1];          // [2,E]
    const int*   bat = (const int*)d_in[2];          // [N]
    const float* W1  = (const float*)d_in[3];        // [7,16]
    const float* b1  = (const float*)d_in[4];        // [16]
    const float* W2  = (const float*)d_in[5];        // [16,16]
    const float* b2  = (const float*)d_in[6];        // [16]
    const float* Wl  = (const float*)d_in[7];        // [16]
    const float* bl  = (const float*)d_in[8];        // [1]
    float* out = (float*)d_out;

    const int N = in_sizes[0] / 7;
    const int E = in_sizes[1] / 2;
    const int G = out_size;                          // 512

    // workspace carve-up (16B aligned slices)
    char* ws = (char*)d_ws;
    size_t off = 0;
    float* deg  = (float*)(ws + off); off += (size_t)N * sizeof(float);          // deg -> dinv
    off = (off + 15) & ~(size_t)15;
    float* bufH = (float*)(ws + off); off += (size_t)N * DH * sizeof(float);     // Hs (scaled gemm out)
    float* bufA = (float*)(ws + off); off += (size_t)N * DH * sizeof(float);     // accum
    float* bufT = (float*)(ws + off); off += (size_t)N * DH * sizeof(float);     // activations
    float* sums = (float*)(ws + off); off += (size_t)NGRAPHS * DH * sizeof(float);
    float* cnts = (float*)(ws + off); off += (size_t)NGRAPHS * sizeof(float);
    (void)ws_size; (void)n_in;

    const int TB = 256;
    dim3 blk(TB);
    int gridNF = (N * DH + TB - 1) / TB;
    int gridE  = (E + TB - 1) / TB;
    int gridN  = (N + TB - 1) / TB;
    int ntiles = (N + 15) / 16;
    int gridG  = (ntiles + (TB / 32) - 1) / (TB / 32);   // 8 waves per block

    // 0) init accumulators / degrees / pool state (fresh every call)
    gcn_init<<<gridNF, blk, 0, stream>>>(deg, bufA, sums, cnts, N);
    // 1) degrees + dinv
    gcn_degree<<<gridE, blk, 0, stream>>>(ei, deg, E);
    gcn_dinv<<<gridN, blk, 0, stream>>>(deg, N);
    // 2) layer 1: Hs = (X @ W1)*dinv (WMMA f32), scatter-add, relu epilogue (+reset accum)
    gcn_gemm_wmma<7><<<gridG, blk, 0, stream>>>(x, W1, deg, bufH, N);
    gcn_edge_agg<<<gridE, blk, 0, stream>>>(ei, bufH, bufA, E);
    gcn_epilogue<<<gridNF, blk, 0, stream>>>(bufH, bufA, deg, b1, bufT, N, 1);
    // 3) layer 2: Hs = (act1 @ W2)*dinv (WMMA f32), scatter-add, relu epilogue
    gcn_gemm_wmma<16><<<gridG, blk, 0, stream>>>(bufT, W2, deg, bufH, N);
    gcn_edge_agg<<<gridE, blk, 0, stream>>>(ei, bufH, bufA, E);
    gcn_epilogue<<<gridNF, blk, 0, stream>>>(bufH, bufA, deg, b2, bufT, N, 0);
    // 4) mean pool + sigmoid head
    gcn_pool<<<gridN, blk, 0, stream>>>(bufT, bat, sums, cnts, N);
    gcn_head<<<(G + TB - 1) / TB, blk, 0, stream>>>(sums, cnts, Wl, bl, out, G);
}